// SinkhornDivergence_35012573397608
// MI455X (gfx1250) — compile-verified
//
#include <hip/hip_runtime.h>
#include <hip/hip_bf16.h>

typedef __attribute__((ext_vector_type(2))) float v2f;
typedef __attribute__((ext_vector_type(8))) float v8f;

// ---------- helpers ----------
static __device__ __forceinline__ float u32_as_f32(unsigned int u) {
    float f; __builtin_memcpy(&f, &u, 4); return f;
}
static __device__ __forceinline__ unsigned int f32_as_u32(float f) {
    unsigned int u; __builtin_memcpy(&u, &f, 4); return u;
}
// round-to-nearest-even f32 -> bf16 (inputs are finite positive, no NaN path needed)
static __device__ __forceinline__ unsigned short f32_to_bf16_rne(float f) {
    unsigned int u = f32_as_u32(f);
    u += 0x7fffu + ((u >> 16) & 1u);
    return (unsigned short)(u >> 16);
}
static __device__ __forceinline__ float wave_reduce_add(float acc) {
    #pragma unroll
    for (int off = 16; off > 0; off >>= 1) acc += __shfl_down(acc, off, 32);
    return acc;
}

// K(c) = max(exp(clip(-c/EPSILON, -50, 0)), 1e-8) with nan/inf guards matching the reference
static __device__ __forceinline__ float cost_guard(float c) {
    c = fmaxf(c, 0.0f);        // clamps negatives, maps NaN -> 0 (nan_to_num(nan=0))
    c = fminf(c, 1.0e6f);      // posinf -> 1e6
    return c;
}
static __device__ __forceinline__ float kernel_from_cost(float c) {
    float s = fmaxf(-c * 100.0f, -50.0f);   // -c/0.01 clipped at -50; upper clip 0 automatic (c>=0)
    return fmaxf(__expf(s), 1.0e-8f);
}

// ---------- row squared-norms: one wave per row ----------
__global__ void sink_rowsq_kernel(const float* __restrict__ A, float* __restrict__ out,
                                  int n, int d) {
    int tid = threadIdx.x, lane = tid & 31, wv = tid >> 5;
    int row = blockIdx.x * 8 + wv;
    if (row >= n) return;
    const float* a = A + (size_t)row * d;
    float acc = 0.0f;
    for (int k = lane * 4; k < d; k += 32 * 4) {
        float4 x = *(const float4*)(a + k);
        acc += x.x * x.x + x.y * x.y + x.z * x.z + x.w * x.w;
    }
    acc = wave_reduce_add(acc);
    if (lane == 0) out[row] = acc;
}

// ---------- build K (bf16): one 16x16 tile per wave via V_WMMA_F32_16X16X4_F32 ----------
__global__ void sink_build_k_kernel(const float* __restrict__ A, const float* __restrict__ B,
                                    const float* __restrict__ sa, const float* __restrict__ sb,
                                    unsigned short* __restrict__ Kout, int n, int m, int d) {
    int tid = threadIdx.x, lane = tid & 31, wv = tid >> 5;
    int tjn = m >> 4;
    int tiles = (n >> 4) * tjn;
    int tile = blockIdx.x * 8 + wv;
    if (tile >= tiles) return;                 // wave-uniform: EXEC stays all-ones for WMMA
    int ti = tile / tjn, tj = tile - ti * tjn;
    int i0 = ti << 4, j0 = tj << 4;

    int r  = lane & 15;            // A: lane -> M row ; B: lane -> N col
    int kh = (lane >> 4) << 1;     // K sub-offset per lane half: 0 or 2
    const float* arow = A + (size_t)(i0 + r) * d + kh;
    const float* brow = B + (size_t)(j0 + r) * d + kh;

    v8f acc = {};
    for (int k = 0; k < d; k += 4) {
        v2f av, bv;
        av.x = arow[k];     av.y = arow[k + 1];   // VGPR0 = K=kh, VGPR1 = K=kh+1
        bv.x = brow[k];     bv.y = brow[k + 1];
        acc = __builtin_amdgcn_wmma_f32_16x16x4_f32(false, av, false, bv,
                                                    (short)0, acc, false, false);
    }

    // D layout: lanes 0-15 -> N=lane, M=vgpr; lanes 16-31 -> N=lane-16, M=8+vgpr
    int nn = lane & 15;
    int mb = (lane >> 4) << 3;
    float sbj = sb[j0 + nn];
    size_t colbase = (size_t)(j0 + nn);
    #pragma unroll
    for (int rr = 0; rr < 8; ++rr) {
        int i = i0 + mb + rr;
        float c = cost_guard(sa[i] + sbj - 2.0f * acc[rr]);
        float kk = kernel_from_cost(c);
        Kout[(size_t)i * m + colbase] = f32_to_bf16_rne(kk);
    }
}

// ---------- u = a / max(K v, eps): one wave per row, 16B bf16 loads ----------
__global__ void sink_kv_u_kernel(const unsigned short* __restrict__ K,
                                 const float* __restrict__ v, float* __restrict__ u,
                                 int n, int m, float aval) {
    int tid = threadIdx.x, lane = tid & 31, wv = tid >> 5;
    int row = blockIdx.x * 8 + wv;
    if (row >= n) return;
    const unsigned short* kr = K + (size_t)row * m;
    float acc = 0.0f;
    for (int base = 0; base < m; base += 256) {
        int j = base + lane * 8;
        uint4  pk = *(const uint4*)(kr + j);
        float4 v0 = *(const float4*)(v + j);
        float4 v1 = *(const float4*)(v + j + 4);
        acc += u32_as_f32(pk.x << 16)          * v0.x;
        acc += u32_as_f32(pk.x & 0xffff0000u)  * v0.y;
        acc += u32_as_f32(pk.y << 16)          * v0.z;
        acc += u32_as_f32(pk.y & 0xffff0000u)  * v0.w;
        acc += u32_as_f32(pk.z << 16)          * v1.x;
        acc += u32_as_f32(pk.z & 0xffff0000u)  * v1.y;
        acc += u32_as_f32(pk.w << 16)          * v1.z;
        acc += u32_as_f32(pk.w & 0xffff0000u)  * v1.w;
    }
    acc = wave_reduce_add(acc);
    if (lane == 0) u[row] = aval / fmaxf(acc, 1.0e-8f);
}

// ---------- K^T u partials: blockIdx.x tiles columns (2048), blockIdx.y tiles rows ----------
__global__ void sink_ktu_partial_kernel(const unsigned short* __restrict__ K,
                                        const float* __restrict__ u,
                                        float* __restrict__ part,
                                        int n, int m, int rows_per) {
    __shared__ float su[256];
    int tid = threadIdx.x;
    int col = blockIdx.x * 2048 + tid * 8;
    int r0  = blockIdx.y * rows_per;
    float acc[8];
    #pragma unroll
    for (int r = 0; r < 8; ++r) acc[r] = 0.0f;

    for (int ib = 0; ib < rows_per; ib += 256) {
        int cnt = rows_per - ib; if (cnt > 256) cnt = 256;
        __syncthreads();
        if (tid < cnt) su[tid] = u[r0 + ib + tid];
        __syncthreads();
        for (int i = 0; i < cnt; ++i) {
            float ui = su[i];
            uint4 pk = *(const uint4*)(K + (size_t)(r0 + ib + i) * m + col);
            acc[0] += u32_as_f32(pk.x << 16)         * ui;
            acc[1] += u32_as_f32(pk.x & 0xffff0000u) * ui;
            acc[2] += u32_as_f32(pk.y << 16)         * ui;
            acc[3] += u32_as_f32(pk.y & 0xffff0000u) * ui;
            acc[4] += u32_as_f32(pk.z << 16)         * ui;
            acc[5] += u32_as_f32(pk.z & 0xffff0000u) * ui;
            acc[6] += u32_as_f32(pk.w << 16)         * ui;
            acc[7] += u32_as_f32(pk.w & 0xffff0000u) * ui;
        }
    }
    float4* p = (float4*)(part + (size_t)blockIdx.y * m + col);
    p[0] = make_float4(acc[0], acc[1], acc[2], acc[3]);
    p[1] = make_float4(acc[4], acc[5], acc[6], acc[7]);
}

// ---------- v = b / max(sum partials, eps) ----------
__global__ void sink_reduce_v_kernel(const float* __restrict__ part, float* __restrict__ v,
                                     int m, int nchunks, float bval) {
    int j = blockIdx.x * 256 + threadIdx.x;
    if (j >= m) return;
    float s = 0.0f;
    for (int c = 0; c < nchunks; ++c) s += part[(size_t)c * m + j];
    v[j] = bval / fmaxf(s, 1.0e-8f);
}

__global__ void sink_init_uv_kernel(float* __restrict__ u, float* __restrict__ v,
                                    int n, int m) {
    int i = blockIdx.x * 256 + threadIdx.x;
    if (i < n) u[i] = 1.0f;
    if (i < m) v[i] = 1.0f;
}

// ---------- final cost: recompute C,K per tile with WMMA; one partial per block ----------
__global__ void sink_cost_tile_kernel(const float* __restrict__ A, const float* __restrict__ B,
                                      const float* __restrict__ sa, const float* __restrict__ sb,
                                      const float* __restrict__ u, const float* __restrict__ v,
                                      float* __restrict__ tpart, int n, int m, int d) {
    __shared__ float wsum[8];
    int tid = threadIdx.x, lane = tid & 31, wv = tid >> 5;
    int tjn = m >> 4;
    int tiles = (n >> 4) * tjn;
    int tile = blockIdx.x * 8 + wv;
    float p = 0.0f;
    if (tile < tiles) {                         // wave-uniform
        int ti = tile / tjn, tj = tile - ti * tjn;
        int i0 = ti << 4, j0 = tj << 4;
        int r  = lane & 15;
        int kh = (lane >> 4) << 1;
        const float* arow = A + (size_t)(i0 + r) * d + kh;
        const float* brow = B + (size_t)(j0 + r) * d + kh;
        v8f acc = {};
        for (int k = 0; k < d; k += 4) {
            v2f av, bv;
            av.x = arow[k]; av.y = arow[k + 1];
            bv.x = brow[k]; bv.y = brow[k + 1];
            acc = __builtin_amdgcn_wmma_f32_16x16x4_f32(false, av, false, bv,
                                                        (short)0, acc, false, false);
        }
        int nn = lane & 15;
        int mb = (lane >> 4) << 3;
        float sbj = sb[j0 + nn];
        float vn  = v[j0 + nn];
        #pragma unroll
        for (int rr = 0; rr < 8; ++rr) {
            int i = i0 + mb + rr;
            float c  = cost_guard(sa[i] + sbj - 2.0f * acc[rr]);
            float kk = kernel_from_cost(c);
            p += u[i] * kk * c;
        }
        p *= vn;
    }
    p = wave_reduce_add(p);
    if (lane == 0) wsum[wv] = p;
    __syncthreads();
    if (tid == 0) {
        float s = 0.0f;
        #pragma unroll
        for (int w = 0; w < 8; ++w) s += wsum[w];
        tpart[blockIdx.x] = s;
    }
}

// ---------- deterministic single-block sum ----------
__global__ void sink_reduce_sum_kernel(const float* __restrict__ arr, int count,
                                       float* __restrict__ out) {
    __shared__ float sh[8];
    int tid = threadIdx.x, lane = tid & 31, wv = tid >> 5;
    float s = 0.0f;
    for (int i = tid; i < count; i += 256) s += arr[i];
    s = wave_reduce_add(s);
    if (lane == 0) sh[wv] = s;
    __syncthreads();
    if (tid == 0) {
        float t = 0.0f;
        #pragma unroll
        for (int w = 0; w < 8; ++w) t += sh[w];
        out[0] = t;
    }
}

__global__ void sink_finalize_kernel(const float* __restrict__ costs, float* __restrict__ out) {
    if (threadIdx.x == 0) {
        float cxy = fmaxf(costs[0], 0.0f);
        float cxx = fmaxf(costs[1], 0.0f);
        float cyy = fmaxf(costs[2], 0.0f);
        float dvr = cxy - 0.5f * (cxx + cyy);
        out[0] = fminf(fmaxf(dvr, 0.0f), 10000.0f);
    }
}

// ---------- host orchestration ----------
extern "C" void kernel_launch(void* const* d_in, const int* in_sizes, int n_in,
                              void* d_out, int out_size, void* d_ws, size_t ws_size,
                              hipStream_t stream) {
    const int D = 128;
    const float* X = (const float*)d_in[0];
    const float* Y = (const float*)d_in[1];
    int n = in_sizes[0] / D;   // 8192
    int m = in_sizes[1] / D;   // 8192

    // workspace layout (256B aligned chunks)
    char* ws = (char*)d_ws;
    size_t off = 0;
    auto take = [&](size_t bytes) { char* p = ws + off; off += (bytes + 255) & ~(size_t)255; return p; };
    size_t nm = (size_t)n * m;
    unsigned short* Kbuf = (unsigned short*)take(nm * 2);
    float* sx    = (float*)take((size_t)n * 4);
    float* sy    = (float*)take((size_t)m * 4);
    float* u     = (float*)take((size_t)n * 4);
    float* v     = (float*)take((size_t)m * 4);
    const int ROW_CHUNKS = 64;
    float* part  = (float*)take((size_t)ROW_CHUNKS * m * 4);
    int tiles    = (n / 16) * (m / 16);
    int tblocks  = (tiles + 7) / 8;
    float* tpart = (float*)take((size_t)tblocks * 4);
    float* costs = (float*)take(4 * 4);

    sink_rowsq_kernel<<<(n + 7) / 8, 256, 0, stream>>>(X, sx, n, D);
    sink_rowsq_kernel<<<(m + 7) / 8, 256, 0, stream>>>(Y, sy, m, D);

    struct Pair { const float* A; const float* B; const float* sa; const float* sb; };
    Pair pairs[3] = { {X, Y, sx, sy}, {X, X, sx, sx}, {Y, Y, sy, sy} };

    float aval = 1.0f / (float)n;
    float bval = 1.0f / (float)m;
    int rows_per = n / ROW_CHUNKS;       // 128
    int col_blocks = m / 2048;           // 4
    int nmax = n > m ? n : m;

    for (int p = 0; p < 3; ++p) {
        const Pair& pr = pairs[p];
        sink_build_k_kernel<<<tblocks, 256, 0, stream>>>(pr.A, pr.B, pr.sa, pr.sb, Kbuf, n, m, D);
        sink_init_uv_kernel<<<(nmax + 255) / 256, 256, 0, stream>>>(u, v, n, m);
        for (int it = 0; it < 100; ++it) {
            sink_kv_u_kernel<<<(n + 7) / 8, 256, 0, stream>>>(Kbuf, v, u, n, m, aval);
            sink_ktu_partial_kernel<<<dim3(col_blocks, ROW_CHUNKS), 256, 0, stream>>>(
                Kbuf, u, part, n, m, rows_per);
            sink_reduce_v_kernel<<<(m + 255) / 256, 256, 0, stream>>>(part, v, m, ROW_CHUNKS, bval);
        }
        sink_cost_tile_kernel<<<tblocks, 256, 0, stream>>>(pr.A, pr.B, pr.sa, pr.sb,
                                                           u, v, tpart, n, m, D);
        sink_reduce_sum_kernel<<<1, 256, 0, stream>>>(tpart, tblocks, costs + p);
    }
    sink_finalize_kernel<<<1, 32, 0, stream>>>(costs, (float*)d_out);
}